// Rpn_69329362092686
// MI455X (gfx1250) — compile-verified
//
#include <hip/hip_runtime.h>
#include <hip/hip_bf16.h>
#include <math.h>

typedef float v2f __attribute__((ext_vector_type(2)));
typedef float v8f __attribute__((ext_vector_type(8)));

#define NPIX     32768      // 8*64*64
#define CIN      512
#define CMID     256
#define NA       9
#define NTOT     294912     // NPIX*9
#define NPRE     12000
#define NPOST    2000
#define NEGF     (-1e30f)
#define CLS_OFF  0
#define REG_OFF  294912
#define PROP_OFF 1474560    // 294912 + 1179648

// ---------------------------------------------------------------- WMMA f32
__device__ inline v8f wmma_f32(v2f a, v2f b, v8f c) {
  // V_WMMA_F32_16X16X4_F32 : D = A(16x4) * B(4x16) + C, full fp32
  return __builtin_amdgcn_wmma_f32_16x16x4_f32(false, a, false, b,
                                               (short)0, c, false, false);
}

// ---------------------------------------------------------------- key utils
__device__ inline unsigned monoU(float s) {
  unsigned u = __float_as_uint(s);
  return (u & 0x80000000u) ? ~u : (u | 0x80000000u);   // monotone: bigger float -> bigger uint
}
__device__ inline unsigned long long makeKey(float s, int i) {
  // unique 64-bit key: score-major, smaller original index wins ties
  return ((unsigned long long)monoU(s) << 32) |
         (unsigned)(0xFFFFFFFFu - (unsigned)i);
}

struct SelState { unsigned long long prefix; unsigned Kr; unsigned pad; };

// ---------------------------------------------------------------- prep kernels
__global__ void prep_w1t_kernel(const float* __restrict__ W1, float* __restrict__ w1t) {
  int idx = blockIdx.x * 256 + threadIdx.x;          // 131072 elements
  if (idx >= CIN * CMID) return;
  int k = idx >> 8;            // 0..511
  int n = idx & 255;           // 0..255
  w1t[n * CIN + k] = W1[idx];  // W1 is [512,256] row-major
}

__global__ void prep_wcr_kernel(const float* __restrict__ Wc, const float* __restrict__ Wr,
                                float* __restrict__ wcrt) {
  int c = blockIdx.x;          // 0..47 (9 cls + 36 reg + 3 zero pad)
  int k = threadIdx.x;         // 0..255
  float v = 0.f;
  if (c < NA)        v = Wc[k * NA + c];
  else if (c < 45)   v = Wr[k * 36 + (c - NA)];
  wcrt[c * CMID + k] = v;
}

__global__ void init_select_kernel(unsigned* hist, SelState* st, unsigned* cnt) {
  int t = threadIdx.x;
  hist[t] = 0u;
  if (t == 0) { st->prefix = 0ull; st->Kr = NPRE; st->pad = 0u; cnt[0] = 0u; }
}

// ---------------------------------------------------------------- fused RPN head
// block = 256 threads = 8 waves; wave w owns pixel rows [base+16w, base+16w+16)
__global__ __launch_bounds__(256)
void head_kernel(const float* __restrict__ feats, const float* __restrict__ w1t,
                 const float* __restrict__ bias1, const float* __restrict__ wcrt,
                 const float* __restrict__ bc, const float* __restrict__ br,
                 float* __restrict__ out)
{
  __shared__ float hTile[8][16][CMID];               // 128 KB: relu(h) staging
  const int wave = threadIdx.x >> 5;
  const int lane = threadIdx.x & 31;
  const int mrow = lane & 15;                        // A layout: M = lane%16
  const int half = lane >> 4;                        // K pair selector
  const int ncol = lane & 15;                        // B/C layout: N = lane%16
  const int rowBase = blockIdx.x * 128 + wave * 16;
  const float* arow = feats + (size_t)(rowBase + mrow) * CIN;

  // ---- GEMM1: h = relu(X*W1 + b1), 4 N-tiles per pass (independent acc chains)
  for (int pass = 0; pass < 4; ++pass) {
    v8f acc[4] = {};
    for (int kb = 0; kb < CIN; kb += 4) {
      v2f a;
      a.x = arow[kb + 2 * half + 0];
      a.y = arow[kb + 2 * half + 1];
#pragma unroll
      for (int t = 0; t < 4; ++t) {
        const int col = (pass * 4 + t) * 16 + ncol;
        const float* bp = w1t + (size_t)col * CIN + kb + 2 * half;
        v2f b; b.x = bp[0]; b.y = bp[1];
        acc[t] = wmma_f32(a, b, acc[t]);
      }
    }
#pragma unroll
    for (int t = 0; t < 4; ++t) {
      const int col = (pass * 4 + t) * 16 + ncol;
      const float bb = bias1[col];
#pragma unroll
      for (int v = 0; v < 8; ++v) {
        float x = acc[t][v] + bb;                    // C layout: M = v + 8*half
        hTile[wave][v + 8 * half][col] = x > 0.f ? x : 0.f;
      }
    }
  }
  __syncthreads();

  // ---- GEMM2: [16x256] * [256x48] (9 cls + 36 reg + pad), A from LDS
  v8f acc2[3] = {};
  for (int kb = 0; kb < CMID; kb += 4) {
    v2f a;
    a.x = hTile[wave][mrow][kb + 2 * half + 0];
    a.y = hTile[wave][mrow][kb + 2 * half + 1];
#pragma unroll
    for (int t = 0; t < 3; ++t) {
      const int col = t * 16 + ncol;
      const float* bp = wcrt + (size_t)col * CMID + kb + 2 * half;
      v2f b; b.x = bp[0]; b.y = bp[1];
      acc2[t] = wmma_f32(a, b, acc2[t]);
    }
  }
#pragma unroll
  for (int t = 0; t < 3; ++t) {
    const int c = t * 16 + ncol;
#pragma unroll
    for (int v = 0; v < 8; ++v) {
      const int p = rowBase + v + 8 * half;          // global pixel index
      float x = acc2[t][v];
      if (c < NA) {
        x += bc[c];
        out[CLS_OFF + p * NA + c] = 1.f / (1.f + expf(-x));   // sigmoid
      } else if (c < 45) {
        x += br[c - NA];
        out[REG_OFF + p * 36 + (c - NA)] = x;
      }
    }
  }
}

// ---------------------------------------------------------------- box decode
__global__ void decode_kernel(const float* __restrict__ out, const float* __restrict__ ancs,
                              const float* __restrict__ aval,
                              float* __restrict__ boxes, float* __restrict__ scores)
{
  int i = blockIdx.x * blockDim.x + threadIdx.x;
  if (i >= NTOT) return;
  float cls = out[CLS_OFF + i];
  scores[i] = (aval[i] > 0.f) ? cls : NEGF;
  float r0 = out[REG_OFF + i * 4 + 0], r1 = out[REG_OFF + i * 4 + 1];
  float r2 = out[REG_OFF + i * 4 + 2], r3 = out[REG_OFF + i * 4 + 3];
  float a0 = ancs[i * 4 + 0], a1 = ancs[i * 4 + 1];
  float a2 = ancs[i * 4 + 2], a3 = ancs[i * 4 + 3];
  float cx = a0 + r0 * a2, cy = a1 + r1 * a3;
  float w  = a2 * expf(r2), h  = a3 * expf(r3);
  boxes[i * 4 + 0] = cx - 0.5f * w;
  boxes[i * 4 + 1] = cy - 0.5f * h;
  boxes[i * 4 + 2] = cx + 0.5f * w;
  boxes[i * 4 + 3] = cy + 0.5f * h;
}

// ---------------------------------------------------------------- radix-select top-12000
__global__ void hist_kernel(const float* __restrict__ scores, unsigned* __restrict__ hist,
                            const SelState* __restrict__ st, int p)
{
  __shared__ unsigned lh[256];
  int t = threadIdx.x;
  lh[t] = 0u;
  __syncthreads();
  int i = blockIdx.x * 256 + t;
  if (i < NTOT) {
    unsigned long long key = makeKey(scores[i], i);
    bool take = true;
    if (p < 7) {
      int hb = 8 * (p + 1);
      take = (key >> hb) == (st->prefix >> hb);
    }
    if (take) atomicAdd(&lh[(unsigned)(key >> (8 * p)) & 255u], 1u);
  }
  __syncthreads();
  if (lh[t]) atomicAdd(&hist[t], lh[t]);
}

__global__ void refine_kernel(unsigned* hist, SelState* st, int p)
{
  __shared__ unsigned h[256];
  int t = threadIdx.x;
  h[t] = hist[t];
  hist[t] = 0u;                                     // reset for next pass
  __syncthreads();
  if (t == 0) {
    unsigned Kr = st->Kr;
    unsigned cum = 0;
    int b = 0;
    for (int bin = 255; bin >= 0; --bin) {
      unsigned c = h[bin];
      if (cum + c >= Kr) { b = bin; st->Kr = Kr - cum; break; }
      cum += c;
    }
    st->prefix |= ((unsigned long long)b) << (8 * p);
  }
}

__global__ void compact_kernel(const float* __restrict__ scores, const float* __restrict__ boxes,
                               const SelState* __restrict__ st, unsigned* __restrict__ cnt,
                               float* __restrict__ topB, float* __restrict__ topS,
                               unsigned* __restrict__ topO)
{
  int i = blockIdx.x * 256 + threadIdx.x;
  if (i >= NTOT) return;
  unsigned long long key = makeKey(scores[i], i);
  if (key >= st->prefix) {                           // unique keys -> exactly NPRE hits
    unsigned pos = atomicAdd(cnt, 1u);
    if (pos < NPRE) {
      topB[pos * 4 + 0] = boxes[i * 4 + 0];
      topB[pos * 4 + 1] = boxes[i * 4 + 1];
      topB[pos * 4 + 2] = boxes[i * 4 + 2];
      topB[pos * 4 + 3] = boxes[i * 4 + 3];
      topS[pos] = scores[i];
      topO[pos] = (unsigned)i;
    }
  }
}

// ---------------------------------------------------------------- greedy NMS, whole set in LDS
__device__ inline unsigned long long shfl_down_u64(unsigned long long v, int off) {
  unsigned lo = (unsigned)v, hi = (unsigned)(v >> 32);
  lo = __shfl_down(lo, off, 32);
  hi = __shfl_down(hi, off, 32);
  return ((unsigned long long)hi << 32) | lo;
}

__global__ __launch_bounds__(1024)
void nms_kernel(const float* __restrict__ topB, const float* __restrict__ topS,
                const unsigned* __restrict__ topO, float* __restrict__ prop)
{
  extern __shared__ unsigned char sm[];              // 288000 B dynamic LDS
  float* X1 = (float*)sm;
  float* Y1 = X1 + NPRE;
  float* X2 = Y1 + NPRE;
  float* Y2 = X2 + NPRE;
  unsigned long long* KEY = (unsigned long long*)(Y2 + NPRE);  // offset 192000, 8-aligned
  __shared__ unsigned long long redK[32];
  __shared__ int redI[32];
  __shared__ unsigned long long sKey;
  __shared__ int sIdx;

  const int tid = threadIdx.x, lane = tid & 31, wid = tid >> 5;

  for (int j = tid; j < NPRE; j += 1024) {
    X1[j] = topB[j * 4 + 0];
    Y1[j] = topB[j * 4 + 1];
    X2[j] = topB[j * 4 + 2];
    Y2[j] = topB[j * 4 + 3];
    KEY[j] = makeKey(topS[j], (int)topO[j]);         // live key; 0 == suppressed
  }
  const unsigned thrHi = monoU(-5e29f);              // validity: score > NEG*0.5
  __syncthreads();

  for (int it = 0; it < NPOST; ++it) {
    // argmax over live keys (deterministic: keys unique)
    unsigned long long bk = 0ull; int bi = 0;
    for (int j = tid; j < NPRE; j += 1024) {
      unsigned long long k = KEY[j];
      if (k > bk) { bk = k; bi = j; }
    }
    for (int off = 16; off > 0; off >>= 1) {
      unsigned long long ok = shfl_down_u64(bk, off);
      int oi = __shfl_down(bi, off, 32);
      if (ok > bk) { bk = ok; bi = oi; }
    }
    if (lane == 0) { redK[wid] = bk; redI[wid] = bi; }
    __syncthreads();
    if (wid == 0) {
      bk = redK[lane]; bi = redI[lane];
      for (int off = 16; off > 0; off >>= 1) {
        unsigned long long ok = shfl_down_u64(bk, off);
        int oi = __shfl_down(bi, off, 32);
        if (ok > bk) { bk = ok; bi = oi; }
      }
      if (lane == 0) { sKey = bk; sIdx = bi; }
    }
    __syncthreads();

    if ((unsigned)(sKey >> 32) <= thrHi) {           // nothing valid left -> zero pad
      for (int q = it * 4 + tid; q < NPOST * 4; q += 1024) prop[q] = 0.f;
      return;
    }
    const int best = sIdx;
    const float b0 = X1[best], c1 = Y1[best], b2 = X2[best], b3 = Y2[best];
    if (tid == 0) {
      prop[it * 4 + 0] = b0; prop[it * 4 + 1] = c1;
      prop[it * 4 + 2] = b2; prop[it * 4 + 3] = b3;
    }
    const float areaB = fmaxf(b2 - b0, 0.f) * fmaxf(b3 - c1, 0.f);
    for (int j = tid; j < NPRE; j += 1024) {
      float xx1 = fmaxf(b0, X1[j]), yy1 = fmaxf(c1, Y1[j]);
      float xx2 = fminf(b2, X2[j]), yy2 = fminf(b3, Y2[j]);
      float inter = fmaxf(xx2 - xx1, 0.f) * fmaxf(yy2 - yy1, 0.f);
      float areaJ = fmaxf(X2[j] - X1[j], 0.f) * fmaxf(Y2[j] - Y1[j], 0.f);
      float iou = inter / (areaB + areaJ - inter + 1e-8f);
      if (iou > 0.7f) KEY[j] = 0ull;
    }
    if (tid == 0) KEY[best] = 0ull;
    __syncthreads();
  }
}

// ---------------------------------------------------------------- launcher
extern "C" void kernel_launch(void* const* d_in, const int* in_sizes, int n_in,
                              void* d_out, int out_size, void* d_ws, size_t ws_size,
                              hipStream_t stream)
{
  const float* feats = (const float*)d_in[0];
  const float* ancs  = (const float*)d_in[1];
  const float* aval  = (const float*)d_in[2];
  const float* W1    = (const float*)d_in[3];
  const float* b1    = (const float*)d_in[4];
  const float* Wc    = (const float*)d_in[5];
  const float* bc    = (const float*)d_in[6];
  const float* Wr    = (const float*)d_in[7];
  const float* br    = (const float*)d_in[8];
  float* out = (float*)d_out;
  char* ws = (char*)d_ws;

  float*    boxes  = (float*)   (ws + 0);        // 294912*4 f32  = 4,718,592 B
  float*    scores = (float*)   (ws + 4718592);  // 294912 f32    = 1,179,648 B
  float*    w1t    = (float*)   (ws + 5898240);  // 256*512 f32   =   524,288 B
  float*    wcrt   = (float*)   (ws + 6422528);  // 48*256 f32    =    49,152 B
  unsigned* hist   = (unsigned*)(ws + 6471680);  // 256 u32
  SelState* st     = (SelState*)(ws + 6472704);
  unsigned* cnt    = (unsigned*)(ws + 6472736);
  float*    topB   = (float*)   (ws + 6472768);  // 12000*4 f32   =   192,000 B
  float*    topS   = (float*)   (ws + 6664768);  // 12000 f32
  unsigned* topO   = (unsigned*)(ws + 6712768);  // 12000 u32 (ends ~6.76 MB)

  prep_w1t_kernel<<<(CIN * CMID + 255) / 256, 256, 0, stream>>>(W1, w1t);
  prep_wcr_kernel<<<48, 256, 0, stream>>>(Wc, Wr, wcrt);
  init_select_kernel<<<1, 256, 0, stream>>>(hist, st, cnt);

  head_kernel<<<NPIX / 128, 256, 0, stream>>>(feats, w1t, b1, wcrt, bc, br, out);
  decode_kernel<<<(NTOT + 255) / 256, 256, 0, stream>>>(out, ancs, aval, boxes, scores);

  for (int p = 7; p >= 0; --p) {
    hist_kernel<<<(NTOT + 255) / 256, 256, 0, stream>>>(scores, hist, st, p);
    refine_kernel<<<1, 256, 0, stream>>>(hist, st, p);
  }
  compact_kernel<<<(NTOT + 255) / 256, 256, 0, stream>>>(scores, boxes, st, cnt,
                                                         topB, topS, topO);

  (void)hipFuncSetAttribute(reinterpret_cast<const void*>(nms_kernel),
                            hipFuncAttributeMaxDynamicSharedMemorySize, 288256);
  nms_kernel<<<1, 1024, 288000, stream>>>(topB, topS, topO, out + PROP_OFF);
}